// FeatureBooster_83837761618430
// MI455X (gfx1250) — compile-verified
//
#include <hip/hip_runtime.h>
#include <math.h>

// FeatureBooster: out = x * sigmoid(2 * (relu(x@W1) @ W2))
// x:[N,256] f32, W1:[256,16] f32, W2:[16,256] f32  -> out:[N,256] f32
//
// Memory-bound (AI ~8 flop/B, ~1.02 GB of mandatory HBM traffic @ 23.3 TB/s
// ~= 44us floor; needed compute ~186 TFLOP/s << f32 WMMA ceiling), so we use
// exact-precision f32 WMMA (V_WMMA_F32_16X16X4_F32): one wave32 per 16-row
// tile, 64 WMMAs for GEMM1 (K=256, two independent accumulator chains for
// ILP), LDS transpose of the 16x16 hidden tile (C/D layout -> A layout),
// 16x4 WMMAs for GEMM2, fused sigmoid gate, non-temporal output stores.

typedef float v2f __attribute__((ext_vector_type(2)));
typedef float v8f __attribute__((ext_vector_type(8)));

#define DIM 256
#define HID 16
#define WAVES_PER_BLOCK 4

__global__ __launch_bounds__(32 * WAVES_PER_BLOCK)
void feature_booster_wmma(const float* __restrict__ x,
                          const float* __restrict__ W1,
                          const float* __restrict__ W2,
                          float* __restrict__ out,
                          int rows)
{
    __shared__ float hbuf[WAVES_PER_BLOCK][HID * HID];  // per-wave 16x16 transpose scratch

    const int wave = threadIdx.x >> 5;
    const int lane = threadIdx.x & 31;
    const int hl   = lane >> 4;   // lane half: 0 -> K={0,1}, 1 -> K={2,3} (A/B frag layout)
    const int nl   = lane & 15;   // index within half (M for A-frags, N for B/C frags)

    const int tile = blockIdx.x * WAVES_PER_BLOCK + wave;
    const int row0 = tile * 16;
    if (row0 >= rows) return;     // whole-wave exit only; EXEC all-ones for WMMA below

    // -------- GEMM1: c1[16x16] = x_tile[16x256] @ W1[256x16] --------
    // A 16x4 f32 layout: lanes 0-15 hold (M=lane, K=0/1) in v0/v1; lanes 16-31 hold K=2/3.
    // Two accumulator chains (even/odd k) to break the WMMA RAW serialization.
    v8f c1a = {0.f, 0.f, 0.f, 0.f, 0.f, 0.f, 0.f, 0.f};
    v8f c1b = {0.f, 0.f, 0.f, 0.f, 0.f, 0.f, 0.f, 0.f};
    const float* xfrag = x + (size_t)(row0 + nl) * DIM + 2 * hl;
    #pragma unroll 4
    for (int k = 0; k < DIM / 4; k += 2) {
        // chain A: step k
        v2f a0 = *(const v2f*)(xfrag + 4 * k);
        v2f b0;
        b0.x = W1[(4 * k + 2 * hl)     * HID + nl];
        b0.y = W1[(4 * k + 2 * hl + 1) * HID + nl];
        c1a = __builtin_amdgcn_wmma_f32_16x16x4_f32(false, a0, false, b0,
                                                    (short)0, c1a, false, false);
        // chain B: step k+1 (independent of chain A)
        v2f a1 = *(const v2f*)(xfrag + 4 * (k + 1));
        v2f b1;
        b1.x = W1[(4 * (k + 1) + 2 * hl)     * HID + nl];
        b1.y = W1[(4 * (k + 1) + 2 * hl + 1) * HID + nl];
        c1b = __builtin_amdgcn_wmma_f32_16x16x4_f32(false, a1, false, b1,
                                                    (short)0, c1b, false, false);
    }

    // Combine chains + ReLU (C/D layout: VGPR v holds M = v + 8*hl, N = nl)
    v8f c1;
    #pragma unroll
    for (int v = 0; v < 8; ++v) c1[v] = fmaxf(c1a[v] + c1b[v], 0.f);

    // -------- transpose hidden tile through per-wave LDS: C layout -> A layout --------
    float* hw = hbuf[wave];
    #pragma unroll
    for (int v = 0; v < 8; ++v)
        hw[(v + 8 * hl) * HID + nl] = c1[v];                  // h[M][k] row-major
    // DS ops are in-order per wave; fence compiler + wait for LDS writes.
    asm volatile("s_wait_dscnt 0" ::: "memory");

    v2f a2[4];                                                // h as A-frags, reused for all 16 n-blocks
    #pragma unroll
    for (int kc = 0; kc < 4; ++kc) {
        const int kk = 4 * kc + 2 * hl;                       // even -> 8B-aligned LDS read
        a2[kc] = *(const v2f*)(hw + nl * HID + kk);           // h[M=nl][kk], h[M=nl][kk+1]
    }

    // -------- GEMM2 + sigmoid gate + store: out[16 x 256] in 16 column blocks --------
    #pragma unroll 2
    for (int n = 0; n < 16; ++n) {
        v8f c2 = {0.f, 0.f, 0.f, 0.f, 0.f, 0.f, 0.f, 0.f};
        #pragma unroll
        for (int kc = 0; kc < 4; ++kc) {
            v2f b;
            b.x = W2[(size_t)(4 * kc + 2 * hl)     * DIM + 16 * n + nl];
            b.y = W2[(size_t)(4 * kc + 2 * hl + 1) * DIM + 16 * n + nl];
            c2 = __builtin_amdgcn_wmma_f32_16x16x4_f32(false, a2[kc], false, b,
                                                       (short)0, c2, false, false);
        }
        #pragma unroll
        for (int v = 0; v < 8; ++v) {
            const int row = row0 + v + 8 * hl;
            const int col = 16 * n + nl;
            const float s  = c2[v];
            const float y  = 1.0f / (1.0f + __expf(-2.0f * s));   // sigmoid(max_out+sum_out)
            const float xv = x[(size_t)row * DIM + col];          // hot in WGP$/L2 (just read)
            __builtin_nontemporal_store(xv * y, &out[(size_t)row * DIM + col]);
        }
    }
}

extern "C" void kernel_launch(void* const* d_in, const int* in_sizes, int n_in,
                              void* d_out, int out_size, void* d_ws, size_t ws_size,
                              hipStream_t stream) {
    const float* x  = (const float*)d_in[0];
    const float* W1 = (const float*)d_in[1];
    const float* W2 = (const float*)d_in[2];
    float* out      = (float*)d_out;

    const int rows   = in_sizes[0] / DIM;                 // 500000
    const int tiles  = (rows + 15) / 16;                  // 31250
    const int blocks = (tiles + WAVES_PER_BLOCK - 1) / WAVES_PER_BLOCK;

    feature_booster_wmma<<<dim3(blocks), dim3(32 * WAVES_PER_BLOCK), 0, stream>>>(
        x, W1, W2, out, rows);
}